// LocalFeaLowDimDynamicFusion_2568390443211
// MI455X (gfx1250) — compile-verified
//
#include <hip/hip_runtime.h>
#include <hip/hip_bf16.h>

typedef __attribute__((ext_vector_type(2))) float v2f;
typedef __attribute__((ext_vector_type(8))) float v8f;

#define NS 8
#define NB 4
#define NA 200
#define NY 20
#define CC 16
#define FH 90
#define FW 160
#define NPTS (NS * NA * NY)          // 32000 points per batch
#define HWSZ (FH * FW)               // 14400
#define EMB_OUT (NB * 256 * NPTS)    // 32768000 floats (first output)

// ---------------------------------------------------------------------------
// embedded_points = W_emb(256x4) @ sp(4xN) + b_emb  -> (B,256,N)
// One wave computes a 16(out-ch) x 16(points) tile with a single
// V_WMMA_F32_16X16X4_F32 (K=4 matches the 4 input coords exactly).
// ---------------------------------------------------------------------------
__global__ __launch_bounds__(256) void embed_kernel(
    const float* __restrict__ sp3d,   // (NS,B,NA,NY,4)
    const float* __restrict__ Wemb,   // (256,4)
    const float* __restrict__ bemb,   // (256,)
    float* __restrict__ out)          // full d_out (embed region at offset 0)
{
    const int lane = threadIdx.x & 31;
    const int wave = blockIdx.x * (blockDim.x >> 5) + (threadIdx.x >> 5);

    const int ptile = wave % (NPTS / 16);
    const int rest  = wave / (NPTS / 16);
    const int ctile = rest & 15;       // 16 tiles of 16 channels = 256
    const int b     = rest >> 4;       // batch

    const int hf = lane >> 4;          // half-wave: 0 -> K{0,1}, 1 -> K{2,3}
    const int lm = lane & 15;
    const int n  = ptile * 16 + lm;    // point index within batch

    // n -> (s, a, y) of the (NS,B,NA,NY,4) input
    const int s = n / (NA * NY);
    const int r = n % (NA * NY);
    const int a = r / NY;
    const int y = r % NY;
    const int in_off = ((((s * NB + b) * NA + a) * NY + y) << 2);

    const float4 p = *(const float4*)(sp3d + in_off);

    // B-matrix (4x16): lane supplies 2 of the 4 K rows for its point
    v2f Bm;
    Bm.x = hf ? p.z : p.x;
    Bm.y = hf ? p.w : p.y;

    // A-matrix (16x4): row = out channel, lane supplies 2 of 4 K entries
    const int row = ctile * 16 + lm;
    v2f Am;
    Am.x = Wemb[row * 4 + 2 * hf + 0];
    Am.y = Wemb[row * 4 + 2 * hf + 1];

    v8f Cm = {};
    Cm = __builtin_amdgcn_wmma_f32_16x16x4_f32(
        /*neg_a=*/false, Am, /*neg_b=*/false, Bm,
        /*c_mod=*/(short)0, Cm, /*reuse_a=*/false, /*reuse_b=*/false);

    // D layout: VGPR j = row (j + 8*hf), col = lane&15
    float* ob = out + (size_t)b * (256 * NPTS)
                    + (size_t)(ctile * 16 + hf * 8) * NPTS + n;
#pragma unroll
    for (int j = 0; j < 8; ++j) {
        const int m = ctile * 16 + hf * 8 + j;
        ob[(size_t)j * NPTS] = Cm[j] + bemb[m];
    }
}

// ---------------------------------------------------------------------------
// local_fea: bilinear grid-sample (zeros padding, align_corners=False) of
// fea(B,16,90,160), then 16x16 conv1d. Per wave: one 16x16 point tile,
// X(16x16) gathered branch-free, D = W_local @ X via 4 chained
// V_WMMA_F32_16X16X4_F32 accumulations.
// Unnormalize collapses to x = u/scale - 0.5, y = v/scale - 0.5.
// ---------------------------------------------------------------------------
__global__ __launch_bounds__(256) void local_kernel(
    const float* __restrict__ fea,    // (B,16,90,160)
    const float* __restrict__ coord,  // (NS,B,NA,NY,2)
    const float* __restrict__ Wl,     // (16,16)
    const float* __restrict__ bl,     // (16,)
    const int*  __restrict__ scale_p, // scalar
    float* __restrict__ out)          // full d_out; local region at EMB_OUT
{
    const int lane = threadIdx.x & 31;
    const int wave = blockIdx.x * (blockDim.x >> 5) + (threadIdx.x >> 5);

    const int ptile = wave % (NPTS / 16);
    const int b     = wave / (NPTS / 16);

    const int hf = lane >> 4;
    const int lm = lane & 15;
    const int n  = ptile * 16 + lm;

    const int s = n / (NA * NY);
    const int r = n % (NA * NY);
    const int a = r / NY;
    const int yy = r % NY;
    const int coff = ((((s * NB + b) * NA + a) * NY + yy) << 1);

    const float inv_scale = 1.0f / (float)scale_p[0];
    const float x = coord[coff + 0] * inv_scale - 0.5f;
    const float y = coord[coff + 1] * inv_scale - 0.5f;

    const float xf = floorf(x), yf = floorf(y);
    const int   x0 = (int)xf,  y0 = (int)yf;
    const float wx1 = x - xf, wx0 = 1.0f - wx1;
    const float wy1 = y - yf, wy0 = 1.0f - wy1;

    const int   xs[2]  = {x0, x0 + 1};
    const int   ys[2]  = {y0, y0 + 1};
    const float wxs[2] = {wx0, wx1};
    const float wys[2] = {wy0, wy1};

    // Branch-free corners: clamped offsets, weight masked by validity.
    float w4[4];
    int   off4[4];
#pragma unroll
    for (int cy = 0; cy < 2; ++cy) {
#pragma unroll
        for (int cx = 0; cx < 2; ++cx) {
            const int xi = xs[cx], yi = ys[cy];
            const bool valid = (xi >= 0) & (xi < FW) & (yi >= 0) & (yi < FH);
            const int xc = min(max(xi, 0), FW - 1);
            const int yc = min(max(yi, 0), FH - 1);
            w4[cy * 2 + cx]   = valid ? (wxs[cx] * wys[cy]) : 0.0f;
            off4[cy * 2 + cx] = yc * FW + xc;
        }
    }

    const float* fb = fea + (size_t)b * CC * HWSZ;

    v8f Cm = {};
#pragma unroll
    for (int k = 0; k < 4; ++k) {
        const int ca = 4 * k + 2 * hf;          // this lane's 2 channels
        const float* f0 = fb + (size_t)ca * HWSZ;
        const float* f1 = fb + (size_t)(ca + 1) * HWSZ;

        v2f Bm;  // X[ch][point] for this K-slice
        Bm.x = w4[0] * f0[off4[0]] + w4[1] * f0[off4[1]]
             + w4[2] * f0[off4[2]] + w4[3] * f0[off4[3]];
        Bm.y = w4[0] * f1[off4[0]] + w4[1] * f1[off4[1]]
             + w4[2] * f1[off4[2]] + w4[3] * f1[off4[3]];

        v2f Am;  // W_local columns ca, ca+1, row = lm
        Am.x = Wl[lm * 16 + ca];
        Am.y = Wl[lm * 16 + ca + 1];

        Cm = __builtin_amdgcn_wmma_f32_16x16x4_f32(
            false, Am, false, Bm, (short)0, Cm, false, false);
    }

    float* ob = out + EMB_OUT + (size_t)b * (CC * NPTS)
                    + (size_t)(hf * 8) * NPTS + n;
#pragma unroll
    for (int j = 0; j < 8; ++j)
        ob[(size_t)j * NPTS] = Cm[j] + bl[hf * 8 + j];
}

extern "C" void kernel_launch(void* const* d_in, const int* in_sizes, int n_in,
                              void* d_out, int out_size, void* d_ws, size_t ws_size,
                              hipStream_t stream) {
    const float* fea   = (const float*)d_in[0];
    const float* sp3d  = (const float*)d_in[1];
    const float* p2d   = (const float*)d_in[2];
    const float* Wl    = (const float*)d_in[3];
    const float* bl    = (const float*)d_in[4];
    const float* Wemb  = (const float*)d_in[5];
    const float* bemb  = (const float*)d_in[6];
    const int*   scale = (const int*)d_in[7];
    float* out = (float*)d_out;

    // embed: 4 batches * 16 channel-tiles * 2000 point-tiles = 128000 waves,
    // 8 waves per 256-thread block -> 16000 blocks.
    embed_kernel<<<16000, 256, 0, stream>>>(sp3d, Wemb, bemb, out);

    // local: 4 * 2000 = 8000 waves -> 1000 blocks.
    local_kernel<<<1000, 256, 0, stream>>>(fea, p2d, Wl, bl, scale, out);
}